// CLAHE_approx_83245056131620
// MI455X (gfx1250) — compile-verified
//
#include <hip/hip_runtime.h>

typedef __attribute__((ext_vector_type(2))) float v2f;
typedef __attribute__((ext_vector_type(8))) float v8f;

#define TILES   8
#define TSIZE   512                  // tile height/width
#define AREA    (TSIZE * TSIZE)      // 262144
#define CLIPF   1228.0f              // int(1.2 * AREA / 256)
#define H_IMG   4096
#define W_IMG   4096
#define C_IMG   3
#define NSEG    (C_IMG * TILES * TILES)   // 192
#define SPLIT   16                   // histogram blocks per tile

// ---------------------------------------------------------------- zero hist
__global__ void clahe_zero_kernel(unsigned int* __restrict__ hist, int n) {
    int i = blockIdx.x * blockDim.x + threadIdx.x;
    if (i < n) hist[i] = 0u;
}

// ---------------------------------------------------------------- histogram
// grid: NSEG*SPLIT blocks of 256 threads; each block: 32 rows x 512 cols.
__global__ void clahe_hist_kernel(const float* __restrict__ img,
                                  unsigned int* __restrict__ hist) {
    __shared__ unsigned int lh[256];
    const int t = threadIdx.x;
    lh[t] = 0u;
    __syncthreads();

    const int bid = blockIdx.x;
    const int s   = bid % SPLIT;
    const int seg = bid / SPLIT;                 // c*64 + ty*8 + tx
    const int tx  = seg % TILES;
    const int ty  = (seg / TILES) % TILES;
    const int c   = seg / (TILES * TILES);

    const int rowsPer = TSIZE / SPLIT;           // 32
    const long baseRow = (long)c * H_IMG + (long)ty * TSIZE + (long)s * rowsPer;

    // 32 rows * 128 float4 = 4096 float4, 256 threads -> 16 iterations
    #pragma unroll 4
    for (int iter = 0; iter < (rowsPer * (TSIZE / 4)) / 256; ++iter) {
        const int linear = iter * 256 + t;
        const int row    = linear >> 7;          // /128
        const int col4   = linear & 127;
        const float4* p  = (const float4*)(img + (baseRow + row) * (long)W_IMG
                                               + (long)tx * TSIZE) + col4;
        __builtin_prefetch((const void*)(p + 256), 0, 0); // next strip ahead
        const float4 px = *p;

        int v0 = (int)(px.x * 255.0f); v0 = v0 < 0 ? 0 : (v0 > 255 ? 255 : v0);
        int v1 = (int)(px.y * 255.0f); v1 = v1 < 0 ? 0 : (v1 > 255 ? 255 : v1);
        int v2 = (int)(px.z * 255.0f); v2 = v2 < 0 ? 0 : (v2 > 255 ? 255 : v2);
        int v3 = (int)(px.w * 255.0f); v3 = v3 < 0 ? 0 : (v3 > 255 ? 255 : v3);
        atomicAdd(&lh[v0], 1u);
        atomicAdd(&lh[v1], 1u);
        atomicAdd(&lh[v2], 1u);
        atomicAdd(&lh[v3], 1u);
    }
    __syncthreads();
    atomicAdd(&hist[seg * 256 + t], lh[t]);
}

// ---------------------------------------------------------------- LUT build
// grid: NSEG blocks of 32 threads (one wave32 each).
// cdf via V_WMMA_F32_16X16X4_F32:  W = H(16x16) x U(upper-tri ones), K=16 as
// 4 chained K=4 WMMAs, then add exclusive prefix of row sums (col 15 of W).
__global__ void clahe_lut_kernel(const unsigned int* __restrict__ hist,
                                 float* __restrict__ lut) {
    __shared__ float h[256];
    const int seg  = blockIdx.x;
    const int lane = threadIdx.x;        // 0..31
    const int m    = lane & 15;
    const int hi   = lane >> 4;

    // clip + excess redistribution (8 bins per lane, strided)
    float excess = 0.0f;
    #pragma unroll
    for (int i = 0; i < 8; ++i) {
        const int b = lane + 32 * i;
        const float x  = (float)hist[seg * 256 + b];
        const float cl = fminf(x, CLIPF);
        excess += x - cl;
        h[b] = cl;
    }
    #pragma unroll
    for (int off = 16; off >= 1; off >>= 1)
        excess += __shfl_xor(excess, off, 32);
    const float add = excess * (1.0f / 256.0f);
    #pragma unroll
    for (int i = 0; i < 8; ++i) h[lane + 32 * i] += add;
    __syncthreads();

    // W = H x U  (within-row inclusive prefix sums)
    v8f acc = {};
    #pragma unroll
    for (int kc = 0; kc < 4; ++kc) {
        const int kb = 4 * kc + 2 * hi;
        v2f a, b;
        // A 16x4 layout: lane -> row m, cols kb, kb+1 of chunk
        a.x = h[m * 16 + kb];
        a.y = h[m * 16 + kb + 1];
        // B 4x16 layout: lane -> col m, rows kb, kb+1; U[k][n] = (k <= n)
        b.x = (kb     <= m) ? 1.0f : 0.0f;
        b.y = (kb + 1 <= m) ? 1.0f : 0.0f;
        acc = __builtin_amdgcn_wmma_f32_16x16x4_f32(
                  false, a, false, b, (short)0, acc, false, false);
    }

    // row sums live in column 15: lane 15 rows 0..7, lane 31 rows 8..15
    float rs[16];
    #pragma unroll
    for (int v = 0; v < 8; ++v) {
        const float w = acc[v];
        rs[v]     = __shfl(w, 15, 32);
        rs[v + 8] = __shfl(w, 31, 32);
    }
    float pre[16];
    pre[0] = 0.0f;
    #pragma unroll
    for (int r = 1; r < 16; ++r) pre[r] = pre[r - 1] + rs[r - 1];

    const float scale = 255.0f / (float)AREA;
    #pragma unroll
    for (int v = 0; v < 8; ++v) {
        const int row = v + 8 * hi;
        const int bin = row * 16 + m;
        const float cdf = acc[v] + pre[row];
        float l = rintf(cdf * scale);            // RNE == jnp.round
        l = fminf(fmaxf(l, 0.0f), 255.0f);
        lut[seg * 256 + bin] = l;
    }
}

// ---------------------------------------------------------------- apply
// grid: (W/1024, H, C), 256 threads, 4 pixels (one float4) per thread.
__global__ void clahe_apply_kernel(const float* __restrict__ img,
                                   const float* __restrict__ lut,
                                   float* __restrict__ out) {
    const int x4 = blockIdx.x * blockDim.x + threadIdx.x;   // 0..1023
    const int y  = blockIdx.y;
    const int c  = blockIdx.z;
    const long rowoff = ((long)c * H_IMG + y) * W_IMG;
    const int  xp     = x4 * 4;

    const float4 px = *(const float4*)(img + rowoff + xp);

    const float fy  = ((float)y + 0.5f) * (1.0f / TSIZE) - 0.5f;
    int y0 = (int)floorf(fy);
    y0 = y0 < 0 ? 0 : (y0 > TILES - 1 ? TILES - 1 : y0);
    float ay = fy - (float)y0;
    ay = ay < 0.0f ? 0.0f : (ay > 1.0f ? 1.0f : ay);
    const int y1 = (y0 + 1 > TILES - 1) ? TILES - 1 : y0 + 1;

    const float* lbase = lut + c * TILES * TILES * 256;
    const float pv[4] = { px.x, px.y, px.z, px.w };
    float res[4];

    #pragma unroll
    for (int j = 0; j < 4; ++j) {
        const int x = xp + j;
        const float fx = ((float)x + 0.5f) * (1.0f / TSIZE) - 0.5f;
        int x0 = (int)floorf(fx);
        x0 = x0 < 0 ? 0 : (x0 > TILES - 1 ? TILES - 1 : x0);
        float ax = fx - (float)x0;
        ax = ax < 0.0f ? 0.0f : (ax > 1.0f ? 1.0f : ax);
        const int x1 = (x0 + 1 > TILES - 1) ? TILES - 1 : x0 + 1;

        int v = (int)(pv[j] * 255.0f);
        v = v < 0 ? 0 : (v > 255 ? 255 : v);

        const float g00 = lbase[(y0 * TILES + x0) * 256 + v];
        const float g01 = lbase[(y0 * TILES + x1) * 256 + v];
        const float g10 = lbase[(y1 * TILES + x0) * 256 + v];
        const float g11 = lbase[(y1 * TILES + x1) * 256 + v];

        const float top = g00 * (1.0f - ax) + g01 * ax;
        const float bot = g10 * (1.0f - ax) + g11 * ax;
        res[j] = (top * (1.0f - ay) + bot * ay) * (1.0f / 255.0f);
    }
    *(float4*)(out + rowoff + xp) = make_float4(res[0], res[1], res[2], res[3]);
}

// ---------------------------------------------------------------- launch
extern "C" void kernel_launch(void* const* d_in, const int* in_sizes, int n_in,
                              void* d_out, int out_size, void* d_ws, size_t ws_size,
                              hipStream_t stream) {
    (void)in_sizes; (void)n_in; (void)out_size; (void)ws_size;
    const float* img = (const float*)d_in[0];
    float* out       = (float*)d_out;

    unsigned int* hist = (unsigned int*)d_ws;                       // 192 KB
    float* lut         = (float*)((char*)d_ws + NSEG * 256 * sizeof(unsigned int));

    const int nh = NSEG * 256;
    clahe_zero_kernel<<<(nh + 255) / 256, 256, 0, stream>>>(hist, nh);

    clahe_hist_kernel<<<NSEG * SPLIT, 256, 0, stream>>>(img, hist);

    clahe_lut_kernel<<<NSEG, 32, 0, stream>>>(hist, lut);

    dim3 grid(W_IMG / (256 * 4), H_IMG, C_IMG);
    clahe_apply_kernel<<<grid, 256, 0, stream>>>(img, lut, out);
}